// BatchNormSPDMean_78134045049543
// MI455X (gfx1250) — compile-verified
//
#include <hip/hip_runtime.h>
#include <hip/hip_bf16.h>

// ---------------------------------------------------------------------------
// Karcher mean of 8192 SPD 64x64 matrices + whitening + SPD bias congruence.
// All matrix functions (sqrt, invsqrt, log, exp) are Newton-Schulz /
// scaling-squaring iterations composed of 64x64x64 f32 matmuls on the CDNA5
// matrix pipe (V_WMMA_F32_16X16X4_F32), all operands LDS-resident.
// The NS "T = 1.5 I - 0.5 Z Y" and Horner "R = I + (A/k) R" updates are fused
// into the WMMA epilogue; buffer copies are replaced by pointer ping-pong.
// ---------------------------------------------------------------------------

#define N    64
#define LD   68            // padded LDS row stride (floats): conflict-free
#define MSZ  (N * LD)
#define TPB  256           // 8 wave32 waves per block
#define ELEMS_PER_BLOCK 16
#define K_SQRT 4           // inverse scaling-squaring depth for logm
#define NS_IT  9           // Newton-Schulz iterations per matrix sqrt
#define EXP_S  6           // scaling-squaring depth for expm

typedef float v2f __attribute__((ext_vector_type(2)));
typedef float v8f __attribute__((ext_vector_type(8)));

#define HAVE_WMMA_F32 __has_builtin(__builtin_amdgcn_wmma_f32_16x16x4_f32)

#define FOR_E(rr, cc) for (int _i = threadIdx.x; _i < N * N; _i += TPB) { \
                        const int rr = _i >> 6, cc = _i & 63;
#define END_E }

// 64x64 (stride LD) f32 matmul using one 256-thread block.
// Wave w owns the 16x32 output strip: tile row ti = (w>>1), tile cols
// tj0 = (w&1)*32 and tj0+16 -> the A fragment is loaded once per K-step and
// feeds two WMMAs. K accumulated in 16 steps of 4 via V_WMMA_F32_16X16X4_F32.
// AFF: C = alpha*(A*B) + delta*I applied on register accumulators.
template <bool AFF>
__device__ __forceinline__ void mm64_t(const float* __restrict__ A,
                                       const float* __restrict__ B,
                                       float* __restrict__ C,
                                       float alpha, float delta) {
#if HAVE_WMMA_F32
  const int lane = threadIdx.x & 31;
  const int wave = threadIdx.x >> 5;
  const int half = lane >> 4;          // 0: lanes 0-15, 1: lanes 16-31
  const int l    = lane & 15;
  const int ti   = (wave >> 1) * 16;   // output tile row offset
  const int tj0  = (wave & 1) * 32;    // first output tile col offset
  v8f acc0 = {}, acc1 = {};
#pragma unroll 4
  for (int kk = 0; kk < 16; ++kk) {
    const int kb = kk * 4 + 2 * half;
    const float* ap = &A[(ti + l) * LD + kb];
    const float* bp = &B[kb * LD + tj0 + l];
    v2f a, b0, b1;
    a.x  = ap[0];       a.y  = ap[1];
    b0.x = bp[0];       b0.y = bp[LD];
    b1.x = bp[16];      b1.y = bp[LD + 16];
    acc0 = __builtin_amdgcn_wmma_f32_16x16x4_f32(false, a, false, b0,
                                                 (short)0, acc0, false, false);
    acc1 = __builtin_amdgcn_wmma_f32_16x16x4_f32(false, a, false, b1,
                                                 (short)0, acc1, false, false);
  }
#pragma unroll
  for (int r = 0; r < 8; ++r) {
    const int row = ti + r + 8 * half;
    float v0 = acc0[r], v1 = acc1[r];
    if (AFF) {
      v0 = alpha * v0 + ((row == tj0 + l)      ? delta : 0.0f);
      v1 = alpha * v1 + ((row == tj0 + 16 + l) ? delta : 0.0f);
    }
    C[row * LD + tj0 + l]      = v0;
    C[row * LD + tj0 + 16 + l] = v1;
  }
#else
  for (int i = threadIdx.x; i < N * N; i += TPB) {
    const int r = i >> 6, c = i & 63;
    float s = 0.0f;
    for (int k = 0; k < N; ++k) s = fmaf(A[r * LD + k], B[k * LD + c], s);
    if (AFF) s = alpha * s + ((r == c) ? delta : 0.0f);
    C[r * LD + c] = s;
  }
#endif
}
__device__ __forceinline__ void mm64(const float* A, const float* B, float* C) {
  mm64_t<false>(A, B, C, 1.0f, 0.0f);
}
__device__ __forceinline__ void mm64_ad(const float* A, const float* B, float* C,
                                        float alpha, float delta) {
  mm64_t<true>(A, B, C, alpha, delta);
}

__device__ __forceinline__ void load_g(float* dst, const float* __restrict__ src) {
  for (int i = threadIdx.x; i < N * N; i += TPB)
    dst[(i >> 6) * LD + (i & 63)] = src[i];
}
__device__ __forceinline__ void store_g(float* __restrict__ dst, const float* src) {
  for (int i = threadIdx.x; i < N * N; i += TPB)
    dst[i] = src[(i >> 6) * LD + (i & 63)];
}
__device__ __forceinline__ void sym_lds(float* A) {
  for (int i = threadIdx.x; i < N * N; i += TPB) {
    const int r = i >> 6, c = i & 63;
    if (r < c) {
      const float v = 0.5f * (A[r * LD + c] + A[c * LD + r]);
      A[r * LD + c] = v;
      A[c * LD + r] = v;
    }
  }
}

__device__ __forceinline__ float frob_norm(const float* A, float* red) {
  float s = 0.0f;
  for (int i = threadIdx.x; i < N * N; i += TPB) {
    const float v = A[(i >> 6) * LD + (i & 63)];
    s += v * v;
  }
  red[threadIdx.x] = s;
  __syncthreads();
  for (int off = TPB >> 1; off > 0; off >>= 1) {
    if (threadIdx.x < off) red[threadIdx.x] += red[threadIdx.x + off];
    __syncthreads();
  }
  const float r0 = sqrtf(red[0]);
  __syncthreads();
  return r0;
}

// Coupled Newton-Schulz with pointer ping-pong; on exit *outY ~ (V/c)^{1/2},
// *outZ ~ (V/c)^{-1/2} (pointers into {V,t1,t2,t4}); returns c.  V destroyed.
__device__ float ns_pair(float* V, float* t1, float* t2, float* t3, float* t4,
                         float* red, int iters, float** outY, float** outZ) {
  const float c = frob_norm(V, red) + 1e-30f;   // Frobenius bounds the spectrum
  const float ic = 1.0f / c;
  float* Y  = t1;
  float* Z  = t2;
  float* T  = t3;
  float* Yn = t4;
  float* Zn = V;                                 // V is dead after the init pass
  FOR_E(r, cc)
    Y[r * LD + cc] = V[r * LD + cc] * ic;
    Z[r * LD + cc] = (r == cc) ? 1.0f : 0.0f;
  END_E
  __syncthreads();
  for (int it = 0; it < iters; ++it) {
    mm64_ad(Z, Y, T, -0.5f, 1.5f);               // T = 1.5 I - 0.5 Z Y (fused)
    __syncthreads();
    mm64(Y, T, Yn);                              // Y' = Y T
    mm64(T, Z, Zn);                              // Z' = T Z
    __syncthreads();
    float* tmp;
    tmp = Y; Y = Yn; Yn = tmp;
    tmp = Z; Z = Zn; Zn = tmp;
  }
  *outY = Y;
  *outZ = Z;
  return c;
}

// ---------------------------------------------------------------------------
// Kernels
// ---------------------------------------------------------------------------

// Block b sums samples [16b, 16b+16) into its private partial slot
// (register-accumulated, fully coalesced, deterministic).
__global__ __launch_bounds__(TPB) void k_sum16(const float* __restrict__ data,
                                               float* __restrict__ part,
                                               int batch) {
  float acc[16];
#pragma unroll
  for (int k = 0; k < 16; ++k) acc[k] = 0.0f;
  const int b0 = blockIdx.x * ELEMS_PER_BLOCK;
  for (int e = 0; e < ELEMS_PER_BLOCK; ++e) {
    const int b = b0 + e;
    if (b >= batch) break;
    const float* src = data + (size_t)b * (N * N);
#pragma unroll
    for (int k = 0; k < 16; ++k) acc[k] += src[threadIdx.x + k * TPB];
  }
  float* dst = part + (size_t)blockIdx.x * (N * N);
#pragma unroll
  for (int k = 0; k < 16; ++k) dst[threadIdx.x + k * TPB] = acc[k];
}

// dst[i] = scale * sum_p part[p][i]   (16 blocks x 256 threads, fixed order)
__global__ __launch_bounds__(TPB) void k_reduce_parts(const float* __restrict__ part,
                                                      int nparts, float scale,
                                                      float* __restrict__ dst) {
  const int i = blockIdx.x * TPB + threadIdx.x;
  if (i < N * N) {
    float s = 0.0f;
    for (int p = 0; p < nparts; ++p) s += part[(size_t)p * (N * N) + i];
    dst[i] = s * scale;
  }
}

__global__ __launch_bounds__(TPB) void k_sqrt_invsqrt(const float* __restrict__ Mg,
                                                      float* __restrict__ Msg,
                                                      float* __restrict__ Misg) {
  __shared__ float B[5][MSZ];
  __shared__ float red[TPB];
  load_g(B[0], Mg);
  __syncthreads();
  sym_lds(B[0]);
  __syncthreads();
  float *Y, *Z;
  const float c = ns_pair(B[0], B[1], B[2], B[3], B[4], red, 16, &Y, &Z);
  const float sc = sqrtf(c), isc = 1.0f / sqrtf(c);
  for (int i = threadIdx.x; i < N * N; i += TPB) {
    const int r = i >> 6, cc = i & 63;
    Msg[i]  = Y[r * LD + cc] * sc;    // M^{1/2}
    Misg[i] = Z[r * LD + cc] * isc;   // M^{-1/2}
  }
}

// Per block: 16 batch samples. W = sym(Mis X Mis); log(W) via 4 NS square
// roots + 4-term Mercator series; accumulate scaled log into a private
// partial slot (deterministic - no float atomics).
__global__ __launch_bounds__(TPB) void k_whiten_log(const float* __restrict__ data,
                                                    const float* __restrict__ Misg,
                                                    float* __restrict__ partial,
                                                    int batch) {
  __shared__ float B[6][MSZ];
  __shared__ float red[TPB];
  load_g(B[0], Misg);                    // Mis stays resident across samples
  __syncthreads();
  const float scale = (float)(1 << K_SQRT) / (float)batch;
  float* slot = partial + (size_t)blockIdx.x * (N * N);
  for (int e = 0; e < ELEMS_PER_BLOCK; ++e) {
    const int b = blockIdx.x * ELEMS_PER_BLOCK + e;
    if (b >= batch) break;
    load_g(B[1], data + (size_t)b * (N * N));
    __syncthreads();
    mm64(B[0], B[1], B[2]);  __syncthreads();   // T1 = Mis*X
    mm64(B[2], B[0], B[1]);  __syncthreads();   // W  = T1*Mis
    sym_lds(B[1]);
    __syncthreads();
    // V = W^{1/2^K_SQRT}; result of each level written back into B[1]
    for (int lsq = 0; lsq < K_SQRT; ++lsq) {
      float *Y, *Z;
      const float c = ns_pair(B[1], B[2], B[3], B[4], B[5], red, NS_IT, &Y, &Z);
      const float sc = sqrtf(c);
      FOR_E(r, cc) B[1][r * LD + cc] = Y[r * LD + cc] * sc; END_E
      __syncthreads();
    }
    // log(V) ~ A - A^2/2 + A^3/3 - A^4/4,  A = V - I  (V near I after roots)
    FOR_E(r, cc)
      B[2][r * LD + cc] = B[1][r * LD + cc] - ((r == cc) ? 1.0f : 0.0f);
    END_E
    __syncthreads();
    mm64(B[2], B[2], B[4]);  __syncthreads();   // A^2
    mm64(B[4], B[2], B[5]);                     // A^3
    mm64(B[4], B[4], B[1]);                     // A^4 (V is dead)
    __syncthreads();
    for (int i = threadIdx.x; i < N * N; i += TPB) {
      const int r = i >> 6, cc = i & 63;
      float v = B[2][r * LD + cc]
              - 0.5f          * B[4][r * LD + cc]
              + (1.0f / 3.0f) * B[5][r * LD + cc]
              - 0.25f         * B[1][r * LD + cc];
      v *= scale;                       // 2^K * logV / batch
      if (e == 0) slot[i] = v; else slot[i] += v;
    }
    __syncthreads();
  }
}

// E = exp(L) by scaling-squaring (8-term Horner + 6 squarings, epilogue-fused,
// ping-pong); M = sym(Ms E Ms).
__global__ __launch_bounds__(TPB) void k_update(const float* __restrict__ Lg,
                                                const float* __restrict__ Msg,
                                                float* __restrict__ Mg) {
  __shared__ float B[4][MSZ];
  float* A = B[0];
  float* R = B[1];
  float* S = B[2];
  for (int i = threadIdx.x; i < N * N; i += TPB)
    A[(i >> 6) * LD + (i & 63)] = Lg[i] * (1.0f / (float)(1 << EXP_S));
  __syncthreads();
  FOR_E(r, c)
    R[r * LD + c] = ((r == c) ? 1.0f : 0.0f) + A[r * LD + c] * (1.0f / 8.0f);
  END_E
  __syncthreads();
  for (int k = 7; k >= 1; --k) {               // R' = I + (A/k) R  (fused)
    mm64_ad(A, R, S, 1.0f / (float)k, 1.0f);
    __syncthreads();
    float* tmp = R; R = S; S = tmp;
  }
  for (int s = 0; s < EXP_S; ++s) {            // square back up
    mm64(R, R, S);
    __syncthreads();
    float* tmp = R; R = S; S = tmp;
  }
  load_g(A, Msg);                              // A := Ms (L/2^s is dead)
  __syncthreads();
  mm64(A, R, B[3]);  __syncthreads();          // Ms*E
  mm64(B[3], A, S);  __syncthreads();          // Ms*E*Ms  (S != B3, != A, != R)
  sym_lds(S);
  __syncthreads();
  store_g(Mg, S);
}

// One-time 64x64 symmetric eig via parallel cyclic Jacobi (round-robin,
// 32 disjoint rotations per step); Bs = U diag(sqrt(softplus(s))) U^T.
__global__ __launch_bounds__(64) void k_bias(const float* __restrict__ C,
                                             float* __restrict__ Bsg) {
  __shared__ float A[N * N];
  __shared__ float U[N * N];
  __shared__ int ord[N];
  __shared__ int pp[N / 2], qq[N / 2];
  __shared__ float cs[N / 2], sn[N / 2];
  __shared__ float w[N];
  const int t = threadIdx.x;
  for (int j = 0; j < N; ++j) {
    A[t * N + j] = 0.5f * (C[t * N + j] + C[j * N + t]);
    U[t * N + j] = (t == j) ? 1.0f : 0.0f;
  }
  ord[t] = t;
  __syncthreads();
  for (int sweep = 0; sweep < 10; ++sweep) {
    for (int step = 0; step < N - 1; ++step) {
      if (t < N / 2) {
        int p = ord[t], q = ord[N - 1 - t];
        if (p > q) { const int tmp = p; p = q; q = tmp; }
        pp[t] = p; qq[t] = q;
        const float app = A[p * N + p], aqq = A[q * N + q], apq = A[p * N + q];
        float c = 1.0f, s = 0.0f;
        if (fabsf(apq) > 1e-12f * (fabsf(app) + fabsf(aqq) + 1e-30f)) {
          const float tau = (aqq - app) / (2.0f * apq);
          const float tt = ((tau >= 0.0f) ? 1.0f : -1.0f)
                           / (fabsf(tau) + sqrtf(1.0f + tau * tau));
          c = rsqrtf(1.0f + tt * tt);
          s = tt * c;
        }
        cs[t] = c; sn[t] = s;
      }
      __syncthreads();
      for (int k = 0; k < N / 2; ++k) {        // column rotations (t = row)
        const int p = pp[k], q = qq[k];
        const float c = cs[k], s = sn[k];
        float x = A[t * N + p], y = A[t * N + q];
        A[t * N + p] = c * x - s * y;  A[t * N + q] = s * x + c * y;
        x = U[t * N + p];  y = U[t * N + q];
        U[t * N + p] = c * x - s * y;  U[t * N + q] = s * x + c * y;
      }
      __syncthreads();
      for (int k = 0; k < N / 2; ++k) {        // row rotations (t = column)
        const int p = pp[k], q = qq[k];
        const float c = cs[k], s = sn[k];
        const float x = A[p * N + t], y = A[q * N + t];
        A[p * N + t] = c * x - s * y;  A[q * N + t] = s * x + c * y;
      }
      __syncthreads();
      if (t == 0) {                  // round-robin: fix ord[0], rotate rest
        const int last = ord[N - 1];
        for (int i = N - 1; i > 1; --i) ord[i] = ord[i - 1];
        ord[1] = last;
      }
      __syncthreads();
    }
  }
  const float d = A[t * N + t];
  w[t] = sqrtf(d > 20.0f ? d : log1pf(expf(d)));   // sqrt(softplus(eig))
  __syncthreads();
  for (int j = 0; j < N; ++j) {
    float s = 0.0f;
    for (int k = 0; k < N; ++k) s += U[t * N + k] * w[k] * U[j * N + k];
    Bsg[t * N + j] = s;
  }
}

// out_b = sym(Bs * sym(G X_b G) * Bs)
__global__ __launch_bounds__(TPB) void k_output(const float* __restrict__ data,
                                                const float* __restrict__ Gg,
                                                const float* __restrict__ Bsg,
                                                float* __restrict__ out) {
  __shared__ float B[5][MSZ];
  load_g(B[0], Gg);
  load_g(B[1], Bsg);
  load_g(B[2], data + (size_t)blockIdx.x * (N * N));
  __syncthreads();
  mm64(B[0], B[2], B[3]);  __syncthreads();
  mm64(B[3], B[0], B[2]);  __syncthreads();
  sym_lds(B[2]);           __syncthreads();
  mm64(B[1], B[2], B[3]);  __syncthreads();
  mm64(B[3], B[1], B[2]);  __syncthreads();
  sym_lds(B[2]);           __syncthreads();
  store_g(out + (size_t)blockIdx.x * (N * N), B[2]);
}

// ---------------------------------------------------------------------------

extern "C" void kernel_launch(void* const* d_in, const int* in_sizes, int n_in,
                              void* d_out, int out_size, void* d_ws, size_t ws_size,
                              hipStream_t stream) {
  (void)n_in; (void)out_size; (void)ws_size;
  const float* data    = (const float*)d_in[0];   // [B,64,64] f32
  const float* covbias = (const float*)d_in[1];   // [64,64]   f32
  float* out = (float*)d_out;
  float* ws  = (float*)d_ws;

  const int batch = in_sizes[0] / (N * N);
  const int nblk  = (batch + ELEMS_PER_BLOCK - 1) / ELEMS_PER_BLOCK;

  float* M    = ws;                 // 4096 f32
  float* Ms   = ws + 1 * N * N;
  float* Mis  = ws + 2 * N * N;     // also G at the end
  float* Bs   = ws + 3 * N * N;
  float* L    = ws + 4 * N * N;
  float* part = ws + 5 * N * N;     // nblk * 4096 f32 partial sums

  // M = arithmetic mean of the batch (deterministic two-stage reduction)
  k_sum16<<<nblk, TPB, 0, stream>>>(data, part, batch);
  k_reduce_parts<<<16, TPB, 0, stream>>>(part, nblk, 1.0f / (float)batch, M);

  for (int it = 0; it < 5; ++it) {
    k_sqrt_invsqrt<<<1, TPB, 0, stream>>>(M, Ms, Mis);
    k_whiten_log<<<nblk, TPB, 0, stream>>>(data, Mis, part, batch);
    k_reduce_parts<<<16, TPB, 0, stream>>>(part, nblk, 1.0f, L);
    k_update<<<1, TPB, 0, stream>>>(L, Ms, M);
  }

  k_sqrt_invsqrt<<<1, TPB, 0, stream>>>(M, Ms, Mis);  // Mis = G = M^{-1/2}
  k_bias<<<1, 64, 0, stream>>>(covbias, Bs);
  k_output<<<batch, TPB, 0, stream>>>(data, Mis, Bs, out);
}